// GNNEncoder_6708738916582
// MI455X (gfx1250) — compile-verified
//
#include <hip/hip_runtime.h>
#include <hip/hip_bf16.h>

#define FDIM 128

typedef float v2f __attribute__((ext_vector_type(2)));
typedef float v8f __attribute__((ext_vector_type(8)));

// ---------------------------------------------------------------------------
// Zero a float buffer (d_ws is poisoned 0xAA; we must initialize scratch).
// ---------------------------------------------------------------------------
__global__ void zero_f32(float* __restrict__ p, long n) {
    long i = (long)blockIdx.x * blockDim.x + threadIdx.x;
    if (i < n) p[i] = 0.0f;
}

// ---------------------------------------------------------------------------
// deg[dst[e]] += 1   (float atomics; deg later inverted in place)
// ---------------------------------------------------------------------------
__global__ void compute_degree(const int* __restrict__ dst,
                               float* __restrict__ deg, int n_edges) {
    int e = blockIdx.x * blockDim.x + threadIdx.x;
    if (e < n_edges) atomicAdd(&deg[dst[e]], 1.0f);
}

// deg[i] = 1 / max(deg[i], 1)   (zero-degree nodes -> agg stays 0, PyG mean)
__global__ void invert_degree(float* __restrict__ deg, int n_nodes) {
    int i = blockIdx.x * blockDim.x + threadIdx.x;
    if (i < n_nodes) deg[i] = 1.0f / fmaxf(deg[i], 1.0f);
}

// ---------------------------------------------------------------------------
// agg[dst[e]][:] += feat[src[e]][:]
// 32 lanes per edge, each lane handles 4 consecutive floats (float4 load,
// 4x global_atomic_add_f32).  feat (51 MB) and agg live in the 192 MB L2.
// ---------------------------------------------------------------------------
__global__ void scatter_add(const float* __restrict__ feat,
                            const int* __restrict__ src,
                            const int* __restrict__ dst,
                            float* __restrict__ agg, int n_edges) {
    long t = (long)blockIdx.x * blockDim.x + threadIdx.x;
    int e = (int)(t >> 5);
    if (e >= n_edges) return;
    int c = ((int)t & 31) * 4;
    const int s = src[e];
    const int d = dst[e];
    const float4 v = *reinterpret_cast<const float4*>(feat + (size_t)s * FDIM + c);
    float* p = agg + (size_t)d * FDIM + c;
    atomicAdd(p + 0, v.x);
    atomicAdd(p + 1, v.y);
    atomicAdd(p + 2, v.z);
    atomicAdd(p + 3, v.w);
}

// ---------------------------------------------------------------------------
// Fused SAGE layer with fp32 WMMA:
//   OUT = act( (AGG .* invdeg) @ Wl^T + bias + X @ Wr^T )
// One wave computes a 16x16 output tile using V_WMMA_F32_16X16X4_F32,
// looping K=128 in steps of 4 for each of the two weight matrices.
//
// Fragment layouts (ISA 7.12.2, wave32):
//   A 16x4 f32 : lane = M (lane&15); VGPR0/1 hold K = kh, kh+1 with
//                kh = 2*(lane>>4)           -> contiguous float2 per lane
//   B 4x16 f32 : lane = N (lane&15); same K split -> B[k][n] = W[n][k],
//                also a contiguous float2 of the row-major weight row
//   C/D 16x16  : row = v + 8*(lane>>4), col = lane&15
// ---------------------------------------------------------------------------
__global__ void sage_layer_wmma(const float* __restrict__ X,
                                const float* __restrict__ AGG,
                                const float* __restrict__ INVD,
                                const float* __restrict__ Wl,   // [128,128] row-major (out,in)
                                const float* __restrict__ Bl,   // [128]
                                const float* __restrict__ Wr,   // [128,128]
                                float* __restrict__ OUT,
                                int n_nodes, int do_relu) {
    const int wave = (int)(((long)blockIdx.x * blockDim.x + threadIdx.x) >> 5);
    const int lane = threadIdx.x & 31;
    const int mtile = wave >> 3;      // 8 column tiles of 16 cover HID=128
    const int ntile = wave & 7;
    const int row0 = mtile * 16;
    if (row0 >= n_nodes) return;      // uniform per wave: EXEC stays all-ones
    const int col0 = ntile * 16;

    const int lm = lane & 15;
    const int kh = (lane >> 4) * 2;   // K sub-offset for this half-wave

    // Clamp defensively (N is a multiple of 16 so normally a no-op);
    // keeps loads in-bounds without perturbing EXEC before the WMMAs.
    const int arow_i = min(row0 + lm, n_nodes - 1);

    const float s = INVD[arow_i];
    const float* arow  = AGG + (size_t)arow_i * FDIM;
    const float* xrow  = X   + (size_t)arow_i * FDIM;
    const float* wlrow = Wl  + (size_t)(col0 + lm) * FDIM;
    const float* wrrow = Wr  + (size_t)(col0 + lm) * FDIM;

    v8f c = {};
    // (mean-aggregated neighbors) @ Wl^T
#pragma unroll
    for (int k = 0; k < FDIM; k += 4) {
        v2f a = *reinterpret_cast<const v2f*>(arow + k + kh);
        a = a * s;
        v2f b = *reinterpret_cast<const v2f*>(wlrow + k + kh);
        c = __builtin_amdgcn_wmma_f32_16x16x4_f32(false, a, false, b,
                                                  (short)0, c, false, false);
    }
    // self features @ Wr^T into the same accumulator
#pragma unroll
    for (int k = 0; k < FDIM; k += 4) {
        v2f a = *reinterpret_cast<const v2f*>(xrow + k + kh);
        v2f b = *reinterpret_cast<const v2f*>(wrrow + k + kh);
        c = __builtin_amdgcn_wmma_f32_16x16x4_f32(false, a, false, b,
                                                  (short)0, c, false, false);
    }

    const float bias = Bl[col0 + lm];
    const int rbase = row0 + (lane >> 4) * 8;
#pragma unroll
    for (int v = 0; v < 8; ++v) {
        int r = rbase + v;
        if (r < n_nodes) {
            float o = c[v] + bias;
            if (do_relu) o = fmaxf(o, 0.0f);
            OUT[(size_t)r * FDIM + col0 + lm] = o;
        }
    }
}

// ---------------------------------------------------------------------------
// Host-side orchestration (everything on `stream`; scratch from d_ws):
//   d_ws layout: agg [N*128] f32 | deg [N] f32 | h [N*128] f32   (~103 MB)
// ---------------------------------------------------------------------------
extern "C" void kernel_launch(void* const* d_in, const int* in_sizes, int n_in,
                              void* d_out, int out_size, void* d_ws, size_t ws_size,
                              hipStream_t stream) {
    const float* x   = (const float*)d_in[0];
    const int*   ei  = (const int*)d_in[1];   // edge_index [2, E]
    const float* W1l = (const float*)d_in[2];
    const float* b1l = (const float*)d_in[3];
    const float* W1r = (const float*)d_in[4];
    const float* W2l = (const float*)d_in[5];
    const float* b2l = (const float*)d_in[6];
    const float* W2r = (const float*)d_in[7];
    float* out = (float*)d_out;

    const int n_nodes = in_sizes[0] / FDIM;
    const int n_edges = in_sizes[1] / 2;
    const int* src = ei;
    const int* dst = ei + n_edges;

    float* agg = (float*)d_ws;
    float* deg = agg + (size_t)n_nodes * FDIM;
    float* h   = deg + n_nodes;

    const int TPB = 256;

    // 1) init agg + deg to zero (contiguous region)
    {
        long n = (long)n_nodes * FDIM + n_nodes;
        zero_f32<<<(int)((n + TPB - 1) / TPB), TPB, 0, stream>>>(agg, n);
    }
    // 2) degree count, 3) invert in place
    compute_degree<<<(n_edges + TPB - 1) / TPB, TPB, 0, stream>>>(dst, deg, n_edges);
    invert_degree<<<(n_nodes + TPB - 1) / TPB, TPB, 0, stream>>>(deg, n_nodes);
    // 4) aggregate layer-1 inputs
    {
        long t = (long)n_edges * 32;
        scatter_add<<<(int)((t + TPB - 1) / TPB), TPB, 0, stream>>>(x, src, dst, agg, n_edges);
    }
    // 5) layer 1: h = relu(mean_agg @ W1l^T + b1l + x @ W1r^T)
    {
        long waves = ((long)(n_nodes + 15) / 16) * 8;
        long t = waves * 32;
        sage_layer_wmma<<<(int)((t + TPB - 1) / TPB), TPB, 0, stream>>>(
            x, agg, deg, W1l, b1l, W1r, h, n_nodes, 1);
    }
    // 6) re-zero agg (deg/invdeg unchanged between layers)
    {
        long n = (long)n_nodes * FDIM;
        zero_f32<<<(int)((n + TPB - 1) / TPB), TPB, 0, stream>>>(agg, n);
    }
    // 7) aggregate layer-2 inputs
    {
        long t = (long)n_edges * 32;
        scatter_add<<<(int)((t + TPB - 1) / TPB), TPB, 0, stream>>>(h, src, dst, agg, n_edges);
    }
    // 8) layer 2 (no relu) -> d_out
    {
        long waves = ((long)(n_nodes + 15) / 16) * 8;
        long t = waves * 32;
        sage_layer_wmma<<<(int)((t + TPB - 1) / TPB), TPB, 0, stream>>>(
            h, agg, deg, W2l, b2l, W2r, out, n_nodes, 0);
    }
}